// Model_2783138807995
// MI455X (gfx1250) — compile-verified
//
#include <hip/hip_runtime.h>

// Problem constants from the reference
#define B_  64
#define S_  256
#define D_  512
#define H_  8
#define HD_ 64

typedef __attribute__((ext_vector_type(2))) float v2f;
typedef __attribute__((ext_vector_type(8))) float v8f;

// ---------------------------------------------------------------------------
// Stage A: wk_eff[d][h] = sum_e Wk[d][h*64+e] * ak[e];  bk_eff[h] = bk_h . ak
// ak = Wa[64:128]  (aq = Wa[0:64] is dead: qa cancels in softmax)
// ---------------------------------------------------------------------------
__global__ void k_wkeff(const float* __restrict__ Wk, const float* __restrict__ Wa,
                        const float* __restrict__ bk,
                        float* __restrict__ wk_eff, float* __restrict__ bk_eff) {
  int idx = blockIdx.x * blockDim.x + threadIdx.x;   // 0..4095
  int d = idx >> 3;
  int h = idx & 7;
  const float* ak = Wa + HD_;
  const float* row = Wk + (size_t)d * D_ + h * HD_;
  float acc = 0.f;
#pragma unroll 8
  for (int e = 0; e < HD_; ++e) acc += row[e] * ak[e];
  wk_eff[d * H_ + h] = acc;
  if (idx < H_) {
    float bacc = 0.f;
    for (int e = 0; e < HD_; ++e) bacc += bk[idx * HD_ + e] * ak[e];
    bk_eff[idx] = bacc;
  }
}

// ---------------------------------------------------------------------------
// Stage B: ka[b][h][j] = X[b,j,:] . wk_eff[:,h] + bk_eff[h]
// One wave per row (b,j); each lane owns 16 consecutive d (float4 loads).
// ---------------------------------------------------------------------------
__global__ void k_ka(const float* __restrict__ X, const float* __restrict__ wk_eff,
                     const float* __restrict__ bk_eff, float* __restrict__ ka) {
  int wave = (blockIdx.x * blockDim.x + threadIdx.x) >> 5;  // row 0..16383
  int lane = threadIdx.x & 31;
  int b = wave >> 8;          // / S_
  int j = wave & (S_ - 1);
  const float4* x4 = (const float4*)(X + (size_t)wave * D_ + lane * 16);
  float acc[H_] = {};
#pragma unroll
  for (int c = 0; c < 4; ++c) {
    float4 xv = x4[c];
    int dbase = lane * 16 + c * 4;
    const float* w0 = wk_eff + dbase * H_;
#pragma unroll
    for (int h = 0; h < H_; ++h)
      acc[h] += xv.x * w0[h] + xv.y * w0[H_ + h] + xv.z * w0[2 * H_ + h] + xv.w * w0[3 * H_ + h];
  }
#pragma unroll
  for (int h = 0; h < H_; ++h) {
    float v = acc[h];
    for (int off = 16; off > 0; off >>= 1) v += __shfl_xor(v, off, 32);
    if (lane == 0) ka[((size_t)b * H_ + h) * S_ + j] = v + bk_eff[h];
  }
}

// ---------------------------------------------------------------------------
// Stage C: dd[b][h][j+1] = desc[b,j, h*64:h*64+64] . ae   (ae = Wa[128:192])
// One wave per (b,j), j in 0..254.
// ---------------------------------------------------------------------------
__global__ void k_ddot(const float* __restrict__ desc, const float* __restrict__ Wa,
                       float* __restrict__ dd) {
  int wave = (blockIdx.x * blockDim.x + threadIdx.x) >> 5;  // 0..16319
  int lane = threadIdx.x & 31;
  int b = wave / (S_ - 1);
  int j = wave % (S_ - 1);
  const float* drow = desc + (size_t)wave * D_;
  const float* ae = Wa + 2 * HD_;
  float ae0 = ae[2 * lane], ae1 = ae[2 * lane + 1];
  float acc[H_];
#pragma unroll
  for (int h = 0; h < H_; ++h) {
    const float* p = drow + h * HD_ + 2 * lane;
    acc[h] = p[0] * ae0 + p[1] * ae1;
  }
#pragma unroll
  for (int h = 0; h < H_; ++h) {
    float v = acc[h];
    for (int off = 16; off > 0; off >>= 1) v += __shfl_xor(v, off, 32);
    if (lane == 0) dd[((size_t)b * H_ + h) * S_ + (j + 1)] = v;
  }
}

// ---------------------------------------------------------------------------
// Stage D: w[b][h][j] = softmax over j=1..255 of (ka + dd);  w[..][0] = 0
// One wave per (b,h).
// ---------------------------------------------------------------------------
__global__ void k_softmax(const float* __restrict__ ka, const float* __restrict__ dd,
                          float* __restrict__ w) {
  int bh = (blockIdx.x * blockDim.x + threadIdx.x) >> 5;  // 0..511
  int lane = threadIdx.x & 31;
  const float* kap = ka + (size_t)bh * S_;
  const float* ddp = dd + (size_t)bh * S_;
  float s[8];
  float mx = -1e30f;
#pragma unroll
  for (int c = 0; c < 8; ++c) {
    int j = lane + c * 32;
    float v = (j == 0) ? -1e30f : (kap[j] + ddp[j]);
    s[c] = v;
    mx = fmaxf(mx, v);
  }
  for (int off = 16; off > 0; off >>= 1) mx = fmaxf(mx, __shfl_xor(mx, off, 32));
  float sum = 0.f;
#pragma unroll
  for (int c = 0; c < 8; ++c) {
    int j = lane + c * 32;
    float e = (j == 0) ? 0.f : expf(s[c] - mx);
    s[c] = e;
    sum += e;
  }
  for (int off = 16; off > 0; off >>= 1) sum += __shfl_xor(sum, off, 32);
  float inv = 1.f / sum;
  float* wp = w + (size_t)bh * S_;
#pragma unroll
  for (int c = 0; c < 8; ++c) wp[lane + c * 32] = s[c] * inv;
}

// ---------------------------------------------------------------------------
// Stage E: xw[b][h][d] = sum_j w[b][h][j] * X[b,j,d]   (streams X once more)
// Block = 256 threads covers 256 d-columns; grid = (2, B). Weights in LDS.
// ---------------------------------------------------------------------------
__global__ void k_xw(const float* __restrict__ X, const float* __restrict__ w,
                     float* __restrict__ xw) {
  __shared__ float wl[H_][S_];
  int b = blockIdx.y;
  int d = blockIdx.x * blockDim.x + threadIdx.x;
  for (int i = threadIdx.x; i < H_ * S_; i += blockDim.x)
    wl[i >> 8][i & (S_ - 1)] = w[(size_t)b * H_ * S_ + i];
  __syncthreads();
  const float* xb = X + (size_t)b * S_ * D_;
  float acc[H_] = {};
  for (int j = 0; j < S_; ++j) {
    float xv = xb[(size_t)j * D_ + d];
#pragma unroll
    for (int h = 0; h < H_; ++h) acc[h] += wl[h][j] * xv;
  }
#pragma unroll
  for (int h = 0; h < H_; ++h) xw[((size_t)b * H_ + h) * D_ + d] = acc[h];
}

// ---------------------------------------------------------------------------
// Stage F: ctx[b][h*64+e] = xw[b][h][:] . Wv[:, h*64+e] + bv   via WMMA fp32
// 128 tiles = h(8) x tm(4 of M=64) x tn(4 of N=64). One wave per 16x16 tile,
// K = 512 in steps of 4 (V_WMMA_F32_16X16X4_F32).
// ---------------------------------------------------------------------------
__global__ void k_ctx(const float* __restrict__ xw, const float* __restrict__ Wv,
                      const float* __restrict__ bv, float* __restrict__ ctx) {
  int wave = (blockIdx.x * blockDim.x + threadIdx.x) >> 5;  // 0..127
  int lane = threadIdx.x & 31;
  int th = wave >> 4;
  int tm = (wave >> 2) & 3;
  int tn = wave & 3;
  int m = lane & 15;
  int klo = (lane >> 4) << 1;
  int col = th * HD_ + tn * 16 + (lane & 15);
  const float* arow = xw + ((size_t)(tm * 16 + m) * H_ + th) * D_;
  v8f c = {};
  for (int kk = 0; kk < D_; kk += 4) {
    v2f a, bm;
    a.x = arow[kk + klo];
    a.y = arow[kk + klo + 1];
    bm.x = Wv[(size_t)(kk + klo) * D_ + col];
    bm.y = Wv[(size_t)(kk + klo + 1) * D_ + col];
    c = __builtin_amdgcn_wmma_f32_16x16x4_f32(false, a, false, bm, (short)0, c, false, false);
  }
  float bias = bv[col];
#pragma unroll
  for (int r = 0; r < 8; ++r) {
    int row = tm * 16 + r + ((lane >> 4) << 3);
    ctx[(size_t)row * D_ + col] = c[r] + bias;
  }
}

// ---------------------------------------------------------------------------
// Stage G: y = ctx @ Wo + bo   (64x512 @ 512x512) via WMMA fp32
// 128 tiles = tm(4) x tn(32). One wave per tile.
// ---------------------------------------------------------------------------
__global__ void k_yproj(const float* __restrict__ ctx, const float* __restrict__ Wo,
                        const float* __restrict__ bo, float* __restrict__ y) {
  int wave = (blockIdx.x * blockDim.x + threadIdx.x) >> 5;  // 0..127
  int lane = threadIdx.x & 31;
  int tm = wave >> 5;
  int tn = wave & 31;
  int m = lane & 15;
  int klo = (lane >> 4) << 1;
  int col = tn * 16 + (lane & 15);
  const float* arow = ctx + (size_t)(tm * 16 + m) * D_;
  v8f c = {};
  for (int kk = 0; kk < D_; kk += 4) {
    v2f a, bm;
    a.x = arow[kk + klo];
    a.y = arow[kk + klo + 1];
    bm.x = Wo[(size_t)(kk + klo) * D_ + col];
    bm.y = Wo[(size_t)(kk + klo + 1) * D_ + col];
    c = __builtin_amdgcn_wmma_f32_16x16x4_f32(false, a, false, bm, (short)0, c, false, false);
  }
  float bias = bo[col];
#pragma unroll
  for (int r = 0; r < 8; ++r) {
    int row = tm * 16 + r + ((lane >> 4) << 3);
    y[(size_t)row * D_ + col] = c[r] + bias;
  }
}

// ---------------------------------------------------------------------------
// Stage H: out[b][s][:] = y[b][:]  (B128 broadcast store, 33.5 MB)
// ---------------------------------------------------------------------------
__global__ void k_bcast(const float* __restrict__ y, float* __restrict__ out) {
  size_t i = (size_t)blockIdx.x * blockDim.x + threadIdx.x;  // over B*S*D/4
  int d4 = (int)(i & (D_ / 4 - 1));
  int b = (int)(i >> 7) >> 8;  // i / 128 / 256
  float4 v = ((const float4*)y)[(size_t)b * (D_ / 4) + d4];
  ((float4*)out)[i] = v;
}

extern "C" void kernel_launch(void* const* d_in, const int* in_sizes, int n_in,
                              void* d_out, int out_size, void* d_ws, size_t ws_size,
                              hipStream_t stream) {
  (void)in_sizes; (void)n_in; (void)out_size; (void)ws_size;
  const float* desc = (const float*)d_in[0];   // (B, S-1, D)
  const float* X    = (const float*)d_in[1];   // (B, S, D) name_value_embeddings
  // d_in[2]=Wq, d_in[3]=bq dead (qa cancels in softmax); d_in[9]=ba dead too.
  const float* Wk = (const float*)d_in[4];
  const float* bk = (const float*)d_in[5];
  const float* Wv = (const float*)d_in[6];
  const float* bv = (const float*)d_in[7];
  const float* Wa = (const float*)d_in[8];
  const float* Wo = (const float*)d_in[10];
  const float* bo = (const float*)d_in[11];
  float* out = (float*)d_out;

  float* ws = (float*)d_ws;
  float* wk_eff = ws;                    // 512*8   = 4096
  float* bk_eff = wk_eff + 4096;         // 32 (pad)
  float* ka     = bk_eff + 32;           // 64*8*256 = 131072
  float* dd     = ka + 131072;           // 131072
  float* w      = dd + 131072;           // 131072
  float* xw     = w  + 131072;           // 64*8*512 = 262144
  float* ctx    = xw + 262144;           // 64*512   = 32768
  float* y      = ctx + 32768;           // 32768
  // total scratch ~ 725k floats ~ 2.9 MB

  k_wkeff  <<<16,   256, 0, stream>>>(Wk, Wa, bk, wk_eff, bk_eff);
  k_ka     <<<2048, 256, 0, stream>>>(X, wk_eff, bk_eff, ka);     // 16384 waves
  k_ddot   <<<2040, 256, 0, stream>>>(desc, Wa, dd);              // 16320 waves
  k_softmax<<<64,   256, 0, stream>>>(ka, dd, w);                 // 512 waves
  dim3 gE(2, B_);
  k_xw     <<<gE,   256, 0, stream>>>(X, w, xw);
  k_ctx    <<<32,   128, 0, stream>>>(xw, Wv, bv, ctx);           // 128 WMMA tiles
  k_yproj  <<<32,   128, 0, stream>>>(ctx, Wo, bo, y);            // 128 WMMA tiles
  k_bcast  <<<8192, 256, 0, stream>>>(y, out);                    // 33.5 MB stores
}